// EmbeddingNetwork_52458730554047
// MI455X (gfx1250) — compile-verified
//
#include <hip/hip_runtime.h>

#define VOCAB   128000
#define EMB_DIM 1024
#define N_ROWS  7
#define SEQ_LEN 8192
#define SPLIT   64                    // blocks per row
#define CHUNK   (SEQ_LEN / SPLIT)     // 128 indices per block

typedef __attribute__((ext_vector_type(2))) float v2f;
typedef __attribute__((ext_vector_type(8))) float v8f;

// ---------------------------------------------------------------- stage 0
__global__ void zero_ws_kernel(float* __restrict__ ws) {
    int i = blockIdx.x * blockDim.x + threadIdx.x;
    if (i < 8 * EMB_DIM) ws[i] = 0.0f;   // 8 rows: rows 0..6 real, row 7 = zero pad
}

// ---------------------------------------------------------------- stage 1
// Fused gather + partial sum. Block = (chunk of 128 seq positions, row r).
// 256 threads; thread t owns dims [4t, 4t+4) -> one float4 (global_load_b128)
// per embedding row, fully coalesced (whole block touches one contiguous 4KB row).
__global__ __launch_bounds__(256) void gather_pool_kernel(
        const int*   __restrict__ ids,
        const float* __restrict__ emb,
        float*       __restrict__ pooled) {
    __shared__ int sidx[CHUNK];
    const int r  = blockIdx.y;
    const int c0 = blockIdx.x * CHUNK;
    const int t  = threadIdx.x;
    if (t < CHUNK) sidx[t] = ids[r * SEQ_LEN + c0 + t];
    __syncthreads();

    const int d = t * 4;
    float4 acc = make_float4(0.f, 0.f, 0.f, 0.f);

    #pragma unroll 4
    for (int i = 0; i < CHUNK; ++i) {
        // look-ahead prefetch of a future row (global_prefetch_b8)
        if (i + 8 < CHUNK) {
            const size_t pbase = ((size_t)sidx[i + 8]) << 10;
            __builtin_prefetch(emb + pbase + d, 0, 1);
        }
        const size_t base = ((size_t)sidx[i]) << 10;   // row * 1024
        const float4 v = *(const float4*)(emb + base + d);
        acc.x += v.x; acc.y += v.y; acc.z += v.z; acc.w += v.w;
    }

    float* dst = pooled + r * EMB_DIM + d;
    atomicAdd(dst + 0, acc.x);
    atomicAdd(dst + 1, acc.y);
    atomicAdd(dst + 2, acc.z);
    atomicAdd(dst + 3, acc.w);
}

// ---------------------------------------------------------------- stage 2
// out[1,1024] = (ratios / SEQ_LEN) @ pooled[8,1024] via V_WMMA_F32_16X16X4_F32.
// One wave32 block per 16-wide N tile; K loop: rows 0..3 then 4..7 (row 7 zero).
//
// 16x4 f32 A layout: lanes 0-15 hold K=0 (v0) / K=1 (v1) for M=lane;
//                    lanes 16-31 hold K=2 (v0) / K=3 (v1) for M=lane-16.
// We put the weight vector in row M=0 (lane 0 and lane 16), rest zero, so
// only D row M=0 (VGPR0, lanes 0-15) is meaningful.
__global__ __launch_bounds__(32) void combine_wmma_kernel(
        const float* __restrict__ ratios,
        const float* __restrict__ pooled,
        float*       __restrict__ out) {
    const int lane = threadIdx.x;            // 0..31
    const int n0   = blockIdx.x * 16;
    const int n    = n0 + (lane & 15);
    const float inv = 1.0f / (float)SEQ_LEN;

    v8f c = {};
    #pragma unroll
    for (int kb = 0; kb < 8; kb += 4) {
        const int kx = kb + (lane < 16 ? 0 : 2);
        const int ky = kb + (lane < 16 ? 1 : 3);

        // A-matrix: weights only in row M=0; branch-free (selects, EXEC stays full)
        const float rx = ratios[kx < N_ROWS ? kx : 0];
        const float ry = ratios[ky < N_ROWS ? ky : 0];
        const float wx = (kx < N_ROWS) ? rx * inv : 0.f;
        const float wy = (ky < N_ROWS) ? ry * inv : 0.f;
        v2f a;
        a.x = ((lane & 15) == 0) ? wx : 0.f;
        a.y = ((lane & 15) == 0) ? wy : 0.f;

        // B-matrix 4x16: VGPR0 = rows K=kb+0 (lanes 0-15), K=kb+2 (lanes 16-31);
        //                VGPR1 = rows K=kb+1,            K=kb+3. Row 7 is zero pad.
        v2f b;
        b.x = pooled[kx * EMB_DIM + n];
        b.y = pooled[ky * EMB_DIM + n];

        c = __builtin_amdgcn_wmma_f32_16x16x4_f32(
                /*neg_a=*/false, a, /*neg_b=*/false, b,
                /*c_mod=*/(short)0, c, /*reuse_a=*/false, /*reuse_b=*/false);
    }

    if (lane < 16) out[n] = c[0];            // D row M=0
}

// ---------------------------------------------------------------- launcher
extern "C" void kernel_launch(void* const* d_in, const int* in_sizes, int n_in,
                              void* d_out, int out_size, void* d_ws, size_t ws_size,
                              hipStream_t stream) {
    const float* ratios = (const float*)d_in[0];   // [7]
    const int*   ids    = (const int*)  d_in[1];   // [7, 8192]
    const float* emb    = (const float*)d_in[2];   // [128000, 1024]
    float*       out    = (float*)d_out;           // [1024]
    float*       pooled = (float*)d_ws;            // [8, 1024] scratch (32 KB)

    zero_ws_kernel<<<(8 * EMB_DIM + 255) / 256, 256, 0, stream>>>(pooled);

    dim3 grid(SPLIT, N_ROWS);
    gather_pool_kernel<<<grid, 256, 0, stream>>>(ids, emb, pooled);

    combine_wmma_kernel<<<EMB_DIM / 16, 32, 0, stream>>>(ratios, pooled, out);
}